// MicroConv_43241730736453
// MI455X (gfx1250) — compile-verified
//
#include <hip/hip_runtime.h>
#include <math.h>

typedef __attribute__((ext_vector_type(2))) float v2f;
typedef __attribute__((ext_vector_type(8))) float v8f;

#define D_IN 256
#define NH 4
#define DF 32
#define HD 128   // NH*DF
#define NEG_SLOPE 0.2f

// ---------------------------------------------------------------------------
// GEMM: C[M,128] = A[M,256] * W[256,128] using V_WMMA_F32_16X16X4_F32.
// One block = 16-row M tile; 8 waves each own one 16-col N tile.
// A 16x4 f32 layout: lane L -> row = L&15, K-half = (L>>4)*2 (2 VGPRs)
// B  4x16 f32 layout: lane L -> col = L&15, K-half = (L>>4)*2 (2 VGPRs)
// C 16x16 f32 layout: lane L -> col = L&15, rows r + (L>>4)*8 in acc[r]
// ---------------------------------------------------------------------------
__global__ __launch_bounds__(256) void gat_gemm_kernel(
    const float* __restrict__ feat_src, const float* __restrict__ feat_dst,
    const float* __restrict__ w_src,    const float* __restrict__ w_dst,
    float* __restrict__ fs, float* __restrict__ fd, int n_src, int n_dst)
{
    const float* A; const float* W; float* C; int M;
    if (blockIdx.y == 0) { A = feat_src; W = w_src; C = fs; M = n_src; }
    else                 { A = feat_dst; W = w_dst; C = fd; M = n_dst; }

    int m0 = blockIdx.x * 16;
    if (m0 >= M) return;
    int lane = threadIdx.x & 31;
    int n0   = (threadIdx.x >> 5) * 16;   // wave id -> N tile
    int t    = lane & 15;                 // A row / B,C col within tile
    int kh   = (lane >> 4) << 1;          // 0 or 2

    int row = m0 + t; if (row >= M) row = M - 1;
    const float* Arow = A + (size_t)row * D_IN;
    const float* Wcol = W + n0 + t;

    v8f acc = {};
    #pragma unroll 8
    for (int k0 = 0; k0 < D_IN; k0 += 4) {
        v2f a = *(const v2f*)(Arow + k0 + kh);
        v2f b;
        b.x = Wcol[(size_t)(k0 + kh)     * HD];
        b.y = Wcol[(size_t)(k0 + kh + 1) * HD];
        acc = __builtin_amdgcn_wmma_f32_16x16x4_f32(
            false, a, false, b, (short)0, acc, false, false);
    }

    int rbase = (lane >> 4) << 3;  // 0 or 8
    float* Cp = C + (size_t)m0 * HD + n0 + t;
    #pragma unroll
    for (int r = 0; r < 8; ++r) {
        int rr = rbase + r;
        if (m0 + rr < M) Cp[(size_t)rr * HD] = acc[r];
    }
}

// ---------------------------------------------------------------------------
// Per-node attention logits: e_src[n,h] = dot(fs[n,h,:], attn[h, D:]),
//                            e_dst[n,h] = dot(fd[n,h,:], attn[h, :D])
// ---------------------------------------------------------------------------
__global__ void gat_logits_kernel(
    const float* __restrict__ fs, const float* __restrict__ fd,
    const float* __restrict__ attn,
    float* __restrict__ e_src, float* __restrict__ e_dst,
    int n_src, int n_dst)
{
    int idx = blockIdx.x * blockDim.x + threadIdx.x;
    int h = idx & (NH - 1);
    int n = idx >> 2;
    if (n < n_src) {
        const float* v = fs + (size_t)n * HD + h * DF;
        const float* a = attn + h * (2 * DF) + DF;   // second half -> src
        float s = 0.f;
        #pragma unroll
        for (int d = 0; d < DF; ++d) s += v[d] * a[d];
        e_src[idx] = s;
    }
    if (n < n_dst) {
        const float* v = fd + (size_t)n * HD + h * DF;
        const float* a = attn + h * (2 * DF);        // first half -> dst
        float s = 0.f;
        #pragma unroll
        for (int d = 0; d < DF; ++d) s += v[d] * a[d];
        e_dst[idx] = s;
    }
}

// ---------------------------------------------------------------------------
__global__ void gat_init_kernel(float* __restrict__ m, float* __restrict__ denom,
                                float* __restrict__ out, int nm, int nout)
{
    int i = blockIdx.x * blockDim.x + threadIdx.x;
    if (i < nout) out[i] = 0.f;
    if (i < nm) { m[i] = -INFINITY; denom[i] = 0.f; }
}

// Sign-split float atomic max (memory initialized to -inf = 0xFF800000).
__device__ __forceinline__ void atomic_max_f32(float* addr, float v)
{
    if (v >= 0.f) atomicMax((int*)addr, __float_as_int(v));
    else          atomicMin((unsigned int*)addr, __float_as_uint(v));
}

__device__ __forceinline__ float leaky(float v)
{
    return v > 0.f ? v : NEG_SLOPE * v;
}

// ---------------------------------------------------------------------------
// Edge pass 1: segment max of leaky-relu logits over dst.
// ---------------------------------------------------------------------------
__global__ void gat_edge_max_kernel(
    const int* __restrict__ src_idx, const int* __restrict__ dst_idx,
    const float* __restrict__ e_src, const float* __restrict__ e_dst,
    float* __restrict__ m, int E)
{
    int e = blockIdx.x * blockDim.x + threadIdx.x;
    if (e >= E) return;
    int s = src_idx[e], d = dst_idx[e];
    float4 es = *(const float4*)(e_src + 4 * (size_t)s);
    float4 ed = *(const float4*)(e_dst + 4 * (size_t)d);
    atomic_max_f32(&m[4 * (size_t)d + 0], leaky(es.x + ed.x));
    atomic_max_f32(&m[4 * (size_t)d + 1], leaky(es.y + ed.y));
    atomic_max_f32(&m[4 * (size_t)d + 2], leaky(es.z + ed.z));
    atomic_max_f32(&m[4 * (size_t)d + 3], leaky(es.w + ed.w));
}

// ---------------------------------------------------------------------------
// Edge pass 2: denom[dst,h] += exp(e - m[dst,h])
// ---------------------------------------------------------------------------
__global__ void gat_edge_sum_kernel(
    const int* __restrict__ src_idx, const int* __restrict__ dst_idx,
    const float* __restrict__ e_src, const float* __restrict__ e_dst,
    const float* __restrict__ m, float* __restrict__ denom, int E)
{
    int e = blockIdx.x * blockDim.x + threadIdx.x;
    if (e >= E) return;
    int s = src_idx[e], d = dst_idx[e];
    float4 es = *(const float4*)(e_src + 4 * (size_t)s);
    float4 ed = *(const float4*)(e_dst + 4 * (size_t)d);
    float4 mm = *(const float4*)(m + 4 * (size_t)d);
    atomicAdd(&denom[4 * (size_t)d + 0], expf(leaky(es.x + ed.x) - mm.x));
    atomicAdd(&denom[4 * (size_t)d + 1], expf(leaky(es.y + ed.y) - mm.y));
    atomicAdd(&denom[4 * (size_t)d + 2], expf(leaky(es.z + ed.z) - mm.z));
    atomicAdd(&denom[4 * (size_t)d + 3], expf(leaky(es.w + ed.w) - mm.w));
}

// ---------------------------------------------------------------------------
// Edge pass 3: one wave32 per edge. Lanes compute a[lane&3] redundantly
// (coalesced), broadcast a[0..3] with __shfl, then 128 feats = 4 coalesced
// loads + 4 f32 atomics per lane into out (L2-resident).
// ---------------------------------------------------------------------------
__global__ __launch_bounds__(256) void gat_edge_agg_kernel(
    const int* __restrict__ src_idx, const int* __restrict__ dst_idx,
    const float* __restrict__ e_src, const float* __restrict__ e_dst,
    const float* __restrict__ m, const float* __restrict__ denom,
    const float* __restrict__ fs, float* __restrict__ out, int E)
{
    int wid = (blockIdx.x << 3) + (threadIdx.x >> 5);
    if (wid >= E) return;
    int lane = threadIdx.x & 31;
    int s = src_idx[wid];
    int d = dst_idx[wid];
    int h = lane & 3;
    float e = e_src[4 * (size_t)s + h] + e_dst[4 * (size_t)d + h];
    e = leaky(e);
    float a = expf(e - m[4 * (size_t)d + h]) / denom[4 * (size_t)d + h];
    float a0 = __shfl(a, 0, 32);
    float a1 = __shfl(a, 1, 32);
    float a2 = __shfl(a, 2, 32);
    float a3 = __shfl(a, 3, 32);
    const float* fsrow = fs + (size_t)s * HD;
    float* orow = out + (size_t)d * HD;
    atomicAdd(orow + lane,      fsrow[lane]      * a0);
    atomicAdd(orow + lane + 32, fsrow[lane + 32] * a1);
    atomicAdd(orow + lane + 64, fsrow[lane + 64] * a2);
    atomicAdd(orow + lane + 96, fsrow[lane + 96] * a3);
}

// ---------------------------------------------------------------------------
__global__ void gat_relu_kernel(float* __restrict__ out, int n)
{
    int i = blockIdx.x * blockDim.x + threadIdx.x;
    if (i < n) out[i] = fmaxf(out[i], 0.f);
}

// ---------------------------------------------------------------------------
extern "C" void kernel_launch(void* const* d_in, const int* in_sizes, int n_in,
                              void* d_out, int out_size, void* d_ws, size_t ws_size,
                              hipStream_t stream)
{
    const float* feat_src = (const float*)d_in[0];
    const float* feat_dst = (const float*)d_in[1];
    const float* w_src    = (const float*)d_in[2];
    const float* w_dst    = (const float*)d_in[3];
    const float* attn     = (const float*)d_in[4];
    const int*   src_idx  = (const int*)d_in[5];
    const int*   dst_idx  = (const int*)d_in[6];
    float* out = (float*)d_out;

    int n_src = in_sizes[0] / D_IN;
    int n_dst = in_sizes[1] / D_IN;
    int E     = in_sizes[5];

    // workspace carve (floats)
    float* ws    = (float*)d_ws;
    float* fs    = ws;  ws += (size_t)n_src * HD;   // 51.2 MB
    float* fd    = ws;  ws += (size_t)n_dst * HD;   // 51.2 MB
    float* e_src = ws;  ws += (size_t)n_src * NH;
    float* e_dst = ws;  ws += (size_t)n_dst * NH;
    float* mx    = ws;  ws += (size_t)n_dst * NH;
    float* denom = ws;  ws += (size_t)n_dst * NH;

    int mmax = n_src > n_dst ? n_src : n_dst;

    dim3 g1((mmax + 15) / 16, 2, 1);
    gat_gemm_kernel<<<g1, 256, 0, stream>>>(feat_src, feat_dst, w_src, w_dst,
                                            fs, fd, n_src, n_dst);

    int nlog = mmax * NH;
    gat_logits_kernel<<<(nlog + 255) / 256, 256, 0, stream>>>(
        fs, fd, attn, e_src, e_dst, n_src, n_dst);

    int nm   = n_dst * NH;
    int nout = n_dst * HD;
    gat_init_kernel<<<(nout + 255) / 256, 256, 0, stream>>>(mx, denom, out, nm, nout);

    gat_edge_max_kernel<<<(E + 255) / 256, 256, 0, stream>>>(
        src_idx, dst_idx, e_src, e_dst, mx, E);

    gat_edge_sum_kernel<<<(E + 255) / 256, 256, 0, stream>>>(
        src_idx, dst_idx, e_src, e_dst, mx, denom, E);

    gat_edge_agg_kernel<<<(E + 7) / 8, 256, 0, stream>>>(
        src_idx, dst_idx, e_src, e_dst, mx, denom, fs, out, E);

    gat_relu_kernel<<<(nout + 255) / 256, 256, 0, stream>>>(out, nout);
}